// BallPointQuery_63256278335590
// MI455X (gfx1250) — compile-verified
//
#include <hip/hip_runtime.h>
#include <cstdint>

typedef __attribute__((ext_vector_type(2))) float v2f;
typedef __attribute__((ext_vector_type(8))) float v8f;

#define BQ_B 4
#define BQ_N 16384
#define BQ_M 2048
#define BQ_MAXS 64
#define BQ_R2 0.04f

// One wave (32 lanes) handles 16 centroids. Distances computed 16x16 at a
// time with V_WMMA_F32_16X16X4_F32:
//   A (16x4): row m = [-2cx, -2cy, -2cz, 1]
//   B (4x16): col n = [ px,  py,  pz, p2]
//   C (16x16): C[m,n] = c2[m]   (broadcast per row)
//   D = A*B + C = c2 + p2 - 2*(c . p) = d2
// Point loads are software-pipelined one tile ahead so the global_load_b96
// latency is hidden behind the WMMA + ballot + bit-drain work of the
// previous tile.
__global__ __launch_bounds__(256) void ballquery_wmma_kernel(
    const float* __restrict__ pcs,        // [B, N, 3]
    const float* __restrict__ centroids,  // [B, M, 3]
    long long*   __restrict__ out)        // [B, M, 64] int64
{
    const int lane = threadIdx.x & 31;
    const int wave = blockIdx.x * (blockDim.x >> 5) + (threadIdx.x >> 5);
    const int tilesPerBatch = BQ_M / 16;                 // 128
    if (wave >= BQ_B * tilesPerBatch) return;            // uniform per wave

    const int b  = wave / tilesPerBatch;
    const int m0 = (wave % tilesPerBatch) * 16;
    const int l  = lane & 15;                            // row / column id
    const bool hi = lane >= 16;
    const int r  = lane & 7;

    // ---- per-wave setup: centroid tile -------------------------------
    const float* cp = centroids + (size_t)(b * BQ_M + m0 + l) * 3;
    const float cx = cp[0], cy = cp[1], cz = cp[2];
    const float c2 = cx * cx + cy * cy + cz * cz;

    // A: lanes 0-15 hold K=0 (vgpr0) / K=1 (vgpr1); lanes 16-31 hold K=2/K=3
    v2f A;
    A.x = hi ? (-2.0f * cz) : (-2.0f * cx);
    A.y = hi ? 1.0f         : (-2.0f * cy);

    // C: VGPR k = c2 of row k (lanes 0-15) / row k+8 (lanes 16-31)
    v8f Cm;
#pragma unroll
    for (int k = 0; k < 8; ++k) {
        float lo = __shfl(c2, k, 32);
        float h8 = __shfl(c2, k + 8, 32);
        Cm[k] = hi ? h8 : lo;
    }

    long long* outp = out + (size_t)(b * BQ_M + m0 + l) * BQ_MAXS;
    const float* pbase = pcs + (size_t)b * BQ_N * 3;

    int cnt = hi ? BQ_MAXS : 0;     // hi lanes never emit -> report "full"
    int firstIdx = BQ_N;

    // ---- software-pipelined scan: 16 points per tile -----------------
    // preload tile 0
    float px, py, pz;
    {
        const float* pp = pbase + (size_t)l * 3;
        px = pp[0]; py = pp[1]; pz = pp[2];
    }

    for (int n0 = 0; n0 < BQ_N; n0 += 16) {
        // issue next tile's load (clamped -> always valid, unconditional)
        const int nnext = (n0 + 16 < BQ_N) ? (n0 + 16) : n0;
        const float* pp = pbase + (size_t)(nnext + l) * 3;
        const float npx = pp[0];
        const float npy = pp[1];
        const float npz = pp[2];

        // consume current tile
        const float p2 = px * px + py * py + pz * pz;
        v2f Bm;
        Bm.x = hi ? pz : px;        // K=2 : K=0
        Bm.y = hi ? p2 : py;        // K=3 : K=1

        v8f D = __builtin_amdgcn_wmma_f32_16x16x4_f32(
            /*neg_a=*/false, A, /*neg_b=*/false, Bm,
            /*c_mod=*/(short)0, Cm, /*reuse_a=*/false, /*reuse_b=*/false);

        // one ballot per D-VGPR: bits[15:0] = row k, bits[31:16] = row k+8
        const unsigned s0 = (unsigned)__ballot(D[0] <= BQ_R2);
        const unsigned s1 = (unsigned)__ballot(D[1] <= BQ_R2);
        const unsigned s2 = (unsigned)__ballot(D[2] <= BQ_R2);
        const unsigned s3 = (unsigned)__ballot(D[3] <= BQ_R2);
        const unsigned s4 = (unsigned)__ballot(D[4] <= BQ_R2);
        const unsigned s5 = (unsigned)__ballot(D[5] <= BQ_R2);
        const unsigned s6 = (unsigned)__ballot(D[6] <= BQ_R2);
        const unsigned s7 = (unsigned)__ballot(D[7] <= BQ_R2);

        // lane l (l<16) grabs its row mask: row l%8 ballot, low/high half
        unsigned bsel = s0;
        bsel = (r == 1) ? s1 : bsel;
        bsel = (r == 2) ? s2 : bsel;
        bsel = (r == 3) ? s3 : bsel;
        bsel = (r == 4) ? s4 : bsel;
        bsel = (r == 5) ? s5 : bsel;
        bsel = (r == 6) ? s6 : bsel;
        bsel = (r == 7) ? s7 : bsel;
        unsigned rowmask = (l < 8) ? (bsel & 0xFFFFu) : (bsel >> 16);
        if (hi || cnt >= BQ_MAXS) rowmask = 0u;

        // drain hits in ascending bit (= ascending point index) order
        while (__any(rowmask != 0u)) {
            if (rowmask) {
                const int bit = __ffs((int)rowmask) - 1;
                const int idx = n0 + bit;
                if (cnt == 0) firstIdx = idx;
                outp[cnt] = (long long)idx;
                ++cnt;
                rowmask &= rowmask - 1u;
                if (cnt >= BQ_MAXS) rowmask = 0u;
            }
        }

        // rotate pipeline registers
        px = npx; py = npy; pz = npz;

        // wave-uniform early exit once all 16 rows hold 64 samples
        if ((unsigned)__ballot(cnt >= BQ_MAXS) == 0xFFFFFFFFu) break;
    }

    // ---- padding: fill remaining slots with first hit (or N) ---------
    if (!hi) {
        const long long pad = (long long)((cnt > 0) ? firstIdx : BQ_N);
        for (int s = cnt; s < BQ_MAXS; ++s) outp[s] = pad;
    }
}

extern "C" void kernel_launch(void* const* d_in, const int* in_sizes, int n_in,
                              void* d_out, int out_size, void* d_ws, size_t ws_size,
                              hipStream_t stream) {
    (void)in_sizes; (void)n_in; (void)out_size; (void)d_ws; (void)ws_size;
    const float* pcs       = (const float*)d_in[0];
    const float* centroids = (const float*)d_in[1];
    long long*   out       = (long long*)d_out;

    // 512 waves total (B*M/16), 8 waves per 256-thread block -> 64 blocks
    const int waves  = (BQ_B * BQ_M) / 16;
    const int blocks = waves / 8;
    ballquery_wmma_kernel<<<blocks, 256, 0, stream>>>(pcs, centroids, out);
}